// CostVolumeBlock_17343077941698
// MI455X (gfx1250) — compile-verified
//
#include <hip/hip_runtime.h>

typedef __attribute__((ext_vector_type(2))) float v2f;
typedef __attribute__((ext_vector_type(8))) float v8f;

#define B_DIM 8
#define H_DIM 192
#define W_DIM 640
#define C_DIM 32
#define SR 2
#define MAXOFF 5                 // 2*SR+1
#define OUTC (C_DIM + MAXOFF)    // 37
#define WAVES 8
#define PIX_PER_BLOCK 128        // 8 waves * 16 pixels
#define TILES_PER_ROW (W_DIM / PIX_PER_BLOCK)  // 5

// One wave computes a 16-pixel tile of one image row using exact-fp32 WMMA.
// D1[j][n] = dot(c1[p0+j], warp[p0-2+n])  (n = j+o,   valid while j+o <= 15)
// D2[j][n] = dot(c1[p0+j], warp[p0+2+n])  (n = j+o-4, covers the rest)
// cost(j,o) = D/32.
__global__ __launch_bounds__(256) void cost_volume_wmma_kernel(
    const float* __restrict__ c1,
    const float* __restrict__ warp,
    float* __restrict__ out)
{
    __shared__ float d1s[WAVES][16][16];
    __shared__ float d2s[WAVES][16][16];

    const int lane = threadIdx.x & 31;
    const int wave = threadIdx.x >> 5;
    const int row  = blockIdx.x / TILES_PER_ROW;   // 0 .. B*H-1
    const int tx   = blockIdx.x % TILES_PER_ROW;
    const int p0   = tx * PIX_PER_BLOCK + wave * 16;

    const float* __restrict__ c1row = c1   + (size_t)row * W_DIM * C_DIM;
    const float* __restrict__ wrow  = warp + (size_t)row * W_DIM * C_DIM;

    // f32 WMMA A/B layout: lane%16 = M (or N) index, lane/16 selects K pair.
    const int m  = lane & 15;
    const int kh = (lane >> 4) * 2;   // 0 or 2 within each K=4 chunk

    const int qa = (p0 - SR) + m;     // B1 rows: p0-2 .. p0+13
    const int qb = (p0 + SR) + m;     // B2 rows: p0+2 .. p0+17

    // Branch-free edge handling: load a clamped (always-valid) address and
    // scale by a 0/1 mask -> exactly reproduces zero padding.
    const float ma = ((unsigned)qa < (unsigned)W_DIM) ? 1.0f : 0.0f;
    const float mb = ((unsigned)qb < (unsigned)W_DIM) ? 1.0f : 0.0f;
    const int qac = qa < 0 ? 0 : (qa > W_DIM - 1 ? W_DIM - 1 : qa);
    const int qbc = qb < 0 ? 0 : (qb > W_DIM - 1 ? W_DIM - 1 : qb);

    const float* __restrict__ aptr  = c1row + (size_t)(p0 + m) * C_DIM + kh;
    const float* __restrict__ b1ptr = wrow  + (size_t)qac * C_DIM + kh;
    const float* __restrict__ b2ptr = wrow  + (size_t)qbc * C_DIM + kh;

    v8f acc1 = {0.f, 0.f, 0.f, 0.f, 0.f, 0.f, 0.f, 0.f};
    v8f acc2 = {0.f, 0.f, 0.f, 0.f, 0.f, 0.f, 0.f, 0.f};

#pragma unroll
    for (int k = 0; k < 8; ++k) {
        v2f a  = *(const v2f*)(aptr  + 4 * k);
        v2f b1 = *(const v2f*)(b1ptr + 4 * k) * ma;
        v2f b2 = *(const v2f*)(b2ptr + 4 * k) * mb;
        // exact fp32 MACs: D += A(16x4) x B(4x16)
        acc1 = __builtin_amdgcn_wmma_f32_16x16x4_f32(
                   false, a, false, b1, (short)0, acc1, false, false);
        acc2 = __builtin_amdgcn_wmma_f32_16x16x4_f32(
                   false, a, false, b2, (short)0, acc2, false, false);
    }

    // Spill D tiles to LDS.  VGPR r of the 16x16 f32 C/D tile holds
    // M = r (lanes 0-15) / M = r+8 (lanes 16-31), N = lane%16.
    const int Mlo = 8 * (lane >> 4);
#pragma unroll
    for (int r = 0; r < 8; ++r) {
        d1s[wave][Mlo + r][m] = acc1[r];
        d2s[wave][Mlo + r][m] = acc2[r];
    }
    __syncthreads();

    float* __restrict__ orow = out + (size_t)row * W_DIM * OUTC;

    // Passthrough of the 32 c1 channels (lane = channel -> coalesced).
    // Output is write-once streaming data: keep it out of cache.
#pragma unroll
    for (int j = 0; j < 16; ++j) {
        __builtin_nontemporal_store(
            c1row[(size_t)(p0 + j) * C_DIM + lane],
            &orow[(size_t)(p0 + j) * OUTC + lane]);
    }

    // Banded gather of the 5 cost channels per pixel (16*5 = 80 values).
    for (int idx = lane; idx < 16 * MAXOFF; idx += 32) {
        const int j = idx / MAXOFF;
        const int o = idx % MAXOFF;
        const int n = j + o;                       // N index in D1
        const float v = (n <= 15) ? d1s[wave][j][n]
                                  : d2s[wave][j][n - 4];
        __builtin_nontemporal_store(
            v * (1.0f / C_DIM),
            &orow[(size_t)(p0 + j) * OUTC + C_DIM + o]);
    }
}

extern "C" void kernel_launch(void* const* d_in, const int* in_sizes, int n_in,
                              void* d_out, int out_size, void* d_ws, size_t ws_size,
                              hipStream_t stream) {
    (void)in_sizes; (void)n_in; (void)out_size; (void)d_ws; (void)ws_size;
    const float* c1   = (const float*)d_in[0];
    const float* warp = (const float*)d_in[1];
    // d_in[2] = search_range (int scalar) — fixed at 2 for this problem.
    float* out = (float*)d_out;

    const int grid = B_DIM * H_DIM * TILES_PER_ROW;  // 1536 * 5 = 7680
    cost_volume_wmma_kernel<<<grid, 256, 0, stream>>>(c1, warp, out);
}